// RangeLoss_73512660239005
// MI455X (gfx1250) — compile-verified
//
#include <hip/hip_runtime.h>
#include <hip/hip_bf16.h>

typedef float v2f __attribute__((ext_vector_type(2)));
typedef float v4f __attribute__((ext_vector_type(4)));
typedef float v8f __attribute__((ext_vector_type(8)));

#define RANGE_HI 1.02f
#define RANGE_LO 0.98f

// ---------------------------------------------------------------------------
// Wave32 reduction using V_WMMA_F32_16X16X4_F32.
// A (16x4 f32, 2 VGPRs/lane): lane L<16 -> A[L][0]=s (v0), A[L][1]=0 (v1)
//                             lane L>=16 -> A[L-16][2]=s, A[L-16][3]=0
// B = all ones (4x16). D[m][n] = s_m + s_{m+16} (independent of n).
// Lane sums its 8 D regs: lanes 0-15 hold sum(u_0..u_7), lanes 16-31 hold
// sum(u_8..u_15). One xor-16 shuffle completes the wave total (in all lanes).
// EXEC must be all 1s at the call site.
// ---------------------------------------------------------------------------
__device__ __forceinline__ float wave_reduce_wmma(float s) {
    v2f a; a.x = s;    a.y = 0.0f;
    v2f b; b.x = 1.0f; b.y = 1.0f;
    v8f c = {};
    v8f d = __builtin_amdgcn_wmma_f32_16x16x4_f32(
        /*neg_a=*/false, a, /*neg_b=*/false, b,
        /*c_mod=*/(short)0, c, /*reuse_a=*/false, /*reuse_b=*/false);
    float r = ((d[0] + d[1]) + (d[2] + d[3])) + ((d[4] + d[5]) + (d[6] + d[7]));
    r += __shfl_xor(r, 16, 32);
    return r;
}

// Block reduction: per-wave WMMA reduce, then thread 0 folds the wave sums.
// Must be called by ALL threads of the block with EXEC full.
__device__ __forceinline__ float block_reduce(float s) {
    __shared__ float sm[32];
    float w = wave_reduce_wmma(s);
    const int lane = threadIdx.x & 31;
    const int wid  = threadIdx.x >> 5;
    if (lane == 0) sm[wid] = w;
    __syncthreads();
    float total = 0.0f;
    if (threadIdx.x == 0) {
        const int nw = (blockDim.x + 31) >> 5;
        #pragma unroll
        for (int i = 0; i < 32; ++i)
            if (i < nw) total += sm[i];
    }
    return total; // valid in thread 0 only
}

// Per-row masked squared error. When the reference substitutes p into t,
// that element's error is exactly zero.
__device__ __forceinline__ float row_err(float p0, float p1, float p2,
                                         float t0, float t1, float t2) {
    const bool c0 = (p0 < 0.1f && t0 == 0.0f) || (p0 > 0.9f && t0 == 1.0f);
    const bool c2 = (p2 < 0.1f && t2 == 0.0f) || (p2 > 0.9f && t2 == 1.0f);
    const bool c1 = (p2 > 0.9f) || ((p1 * RANGE_HI > t1) && (p1 * RANGE_LO < t1));
    const float d0 = p0 - t0, d1 = p1 - t1, d2 = p2 - t2;
    float e = c0 ? 0.0f : d0 * d0;
    e += c1 ? 0.0f : d1 * d1;
    e += c2 ? 0.0f : d2 * d2;
    return e;
}

// Stage 1: each thread handles 4 rows = 3 x float4 per array (48B, aligned),
// streamed with non-temporal 128-bit loads. One partial per block.
__global__ void rl_partial_kernel(const float* __restrict__ preds,
                                  const float* __restrict__ target,
                                  float* __restrict__ partial,
                                  long long ngroups) {
    const long long g = (long long)blockIdx.x * blockDim.x + threadIdx.x;
    float s = 0.0f;
    if (g < ngroups) {
        const v4f* P = (const v4f*)(preds  + 12ll * g);
        const v4f* T = (const v4f*)(target + 12ll * g);
        const v4f P0 = __builtin_nontemporal_load(P + 0);
        const v4f P1 = __builtin_nontemporal_load(P + 1);
        const v4f P2 = __builtin_nontemporal_load(P + 2);
        const v4f T0 = __builtin_nontemporal_load(T + 0);
        const v4f T1 = __builtin_nontemporal_load(T + 1);
        const v4f T2 = __builtin_nontemporal_load(T + 2);
        s  = row_err(P0.x, P0.y, P0.z, T0.x, T0.y, T0.z);
        s += row_err(P0.w, P1.x, P1.y, T0.w, T1.x, T1.y);
        s += row_err(P1.z, P1.w, P2.x, T1.z, T1.w, T2.x);
        s += row_err(P2.y, P2.z, P2.w, T2.y, T2.z, T2.w);
    }
    const float bs = block_reduce(s);
    if (threadIdx.x == 0) partial[blockIdx.x] = bs;
}

// Stage 2: single block folds the block partials (plus any tail rows) and
// writes mean = sum / (3N).
__global__ void rl_final_kernel(const float* __restrict__ partial, int nparts,
                                const float* __restrict__ preds,
                                const float* __restrict__ target,
                                long long tail_row0, int tail_rows,
                                float inv_count, float* __restrict__ out) {
    float s = 0.0f;
    for (int i = threadIdx.x; i < nparts; i += blockDim.x)
        s += partial[i];
    float total = block_reduce(s);
    if (threadIdx.x == 0) {
        for (int r = 0; r < tail_rows; ++r) {
            const long long base = 3ll * (tail_row0 + r);
            total += row_err(preds[base], preds[base + 1], preds[base + 2],
                             target[base], target[base + 1], target[base + 2]);
        }
        out[0] = total * inv_count;
    }
}

extern "C" void kernel_launch(void* const* d_in, const int* in_sizes, int n_in,
                              void* d_out, int out_size, void* d_ws, size_t ws_size,
                              hipStream_t stream) {
    const float* preds  = (const float*)d_in[0];
    const float* target = (const float*)d_in[1];
    float* out = (float*)d_out;
    float* partial = (float*)d_ws;

    const long long N = (long long)in_sizes[0] / 3; // rows
    const long long ngroups = N / 4;                // 4 rows per thread
    const int  tail_rows = (int)(N - ngroups * 4);
    const long long tail_row0 = ngroups * 4;

    const int block = 256;
    const int nblk = (int)((ngroups + block - 1) / block); // 8192 for N=2^23

    rl_partial_kernel<<<nblk, block, 0, stream>>>(preds, target, partial, ngroups);

    const float inv_count = (float)(1.0 / (3.0 * (double)N));
    rl_final_kernel<<<1, block, 0, stream>>>(partial, nblk, preds, target,
                                             tail_row0, tail_rows, inv_count, out);
}